// SAGEEncoder_32959579030041
// MI455X (gfx1250) — compile-verified
//
#include <hip/hip_runtime.h>
#include <stddef.h>

typedef __attribute__((ext_vector_type(2))) float v2f;
typedef __attribute__((ext_vector_type(8))) float v8f;

#define N_NODES 100000
#define N_EDGES 1600000
#define IN_CH   128      // layer-1 input dim == HID, so feature rows are always 128 wide
#define HID     128
#define OUT_CH  64

// workspace layout in floats
#define CNT_OFF  0                            // N_NODES
#define AGG_OFF  131072                       // N_NODES*128 (512KB-aligned start)
#define H_OFF    (AGG_OFF + N_NODES * 128)    // N_NODES*128
// total = H_OFF + N_NODES*128 = 25,731,072 floats ~= 103 MB

// ---------------------------------------------------------------- zero fill
__global__ __launch_bounds__(256) void zero_f4(float4* __restrict__ p, int n4) {
    int i = blockIdx.x * 256 + threadIdx.x;
    if (i < n4) p[i] = make_float4(0.f, 0.f, 0.f, 0.f);
}

// ---------------------------------------------------------------- degree histogram
__global__ __launch_bounds__(256) void degree_kernel(const int* __restrict__ dst,
                                                     float* __restrict__ cnt, int n_edges) {
    int e = blockIdx.x * 256 + threadIdx.x;
    if (e < n_edges) atomicAdd(&cnt[dst[e]], 1.0f);
}

// ---------------------------------------------------------------- edge scatter-add
// one wave32 per edge; lane reads float4 of the 128-wide source row (coalesced 512B)
// and atomically adds into the destination row (L2-resident atomics).
__global__ __launch_bounds__(256) void scatter_kernel(const float* __restrict__ feat,
                                                      const int* __restrict__ src,
                                                      const int* __restrict__ dst,
                                                      float* __restrict__ agg, int n_edges) {
    int e = blockIdx.x * 8 + (threadIdx.x >> 5);
    if (e >= n_edges) return;
    int lane = threadIdx.x & 31;
    int s = src[e];
    int d = dst[e];
    const float4 v = *(const float4*)(feat + (size_t)s * 128 + lane * 4);
    float* p = agg + (size_t)d * 128 + lane * 4;
    atomicAdd(p + 0, v.x);
    atomicAdd(p + 1, v.y);
    atomicAdd(p + 2, v.z);
    atomicAdd(p + 3, v.w);
}

// ---------------------------------------------------------------- fused SAGE GEMM
// out[16n x 16c tile] = relu?( (agg/deg) @ W_l^T + b_l + feat @ W_r^T )
// One wave per (node_tile, out_tile). K-loop in steps of 4 using V_WMMA_F32_16X16X4_F32.
//
// A (16x4 f32) lane layout (ISA 7.12.2): lane l holds row m = l&15;
//   lanes 0-15 carry K = k,k+1 ; lanes 16-31 carry K = k+2,k+3 -> one float2 per lane.
// B (4x16 f32) mirrored: lane l holds col n = l&15, same K-pair split; since
//   B[k][n] = W[n][k], that is a contiguous float2 from W row (out_base + n).
// C/D (16x16 f32): VGPR i -> row (lane<16 ? i : i+8), col = lane&15.
__global__ __launch_bounds__(128) void sage_gemm_kernel(const float* __restrict__ feat,
                                                        const float* __restrict__ agg,
                                                        const float* __restrict__ cnt,
                                                        const float* __restrict__ W_l,
                                                        const float* __restrict__ b_l,
                                                        const float* __restrict__ W_r,
                                                        float* __restrict__ out,
                                                        int n_out, int do_relu) {
    const int lane      = threadIdx.x & 31;
    const int wave      = threadIdx.x >> 5;
    const int node_base = blockIdx.x * 16;
    const int out_base  = (blockIdx.y * 4 + wave) * 16;

    const int m     = lane & 15;        // A row within tile / B col within tile
    const int khalf = (lane >> 4) << 1; // 0 for lanes 0-15, 2 for lanes 16-31

    const int   row = node_base + m;
    const float inv = 1.0f / fmaxf(cnt[row], 1.0f); // mean; isolated nodes -> agg==0 anyway

    const float* __restrict__ Aagg = agg  + (size_t)row * 128;
    const float* __restrict__ Ax   = feat + (size_t)row * 128;
    const float* __restrict__ Bl   = W_l  + (size_t)(out_base + m) * 128;
    const float* __restrict__ Br   = W_r  + (size_t)(out_base + m) * 128;

    v8f acc = {};
#pragma unroll
    for (int k = 0; k < 128; k += 4) {
        v2f a1 = *(const v2f*)(Aagg + k + khalf);
        a1 *= inv;                                   // fold 1/deg into A rows
        v2f a2 = *(const v2f*)(Ax + k + khalf);
        v2f bl = *(const v2f*)(Bl + k + khalf);
        v2f br = *(const v2f*)(Br + k + khalf);
        acc = __builtin_amdgcn_wmma_f32_16x16x4_f32(false, a1, false, bl,
                                                    (short)0, acc, false, false);
        acc = __builtin_amdgcn_wmma_f32_16x16x4_f32(false, a2, false, br,
                                                    (short)0, acc, false, false);
    }

    const float bias = b_l[out_base + m];
#pragma unroll
    for (int i = 0; i < 8; ++i) {
        float v = acc[i] + bias;
        if (do_relu) v = fmaxf(v, 0.0f);
        const int mm = (lane < 16) ? i : (i + 8);
        out[(size_t)(node_base + mm) * n_out + out_base + m] = v;
    }
}

// ---------------------------------------------------------------- launch
extern "C" void kernel_launch(void* const* d_in, const int* in_sizes, int n_in,
                              void* d_out, int out_size, void* d_ws, size_t ws_size,
                              hipStream_t stream) {
    const float* x    = (const float*)d_in[0];
    const int*   ei   = (const int*)d_in[1];     // [2, N_EDGES]
    const float* W1_l = (const float*)d_in[2];
    const float* b1_l = (const float*)d_in[3];
    const float* W1_r = (const float*)d_in[4];
    const float* W2_l = (const float*)d_in[5];
    const float* b2_l = (const float*)d_in[6];
    const float* W2_r = (const float*)d_in[7];
    float*       out  = (float*)d_out;

    const int* src = ei;
    const int* dst = ei + N_EDGES;

    float* ws  = (float*)d_ws;
    float* cnt = ws + CNT_OFF;
    float* agg = ws + AGG_OFF;
    float* h   = ws + H_OFF;

    const int zero_all_f4 = (AGG_OFF + N_NODES * 128) / 4;  // cnt + agg region
    const int zero_agg_f4 = (N_NODES * 128) / 4;

    // ---- layer 1 ----
    zero_f4<<<(zero_all_f4 + 255) / 256, 256, 0, stream>>>((float4*)ws, zero_all_f4);
    degree_kernel<<<(N_EDGES + 255) / 256, 256, 0, stream>>>(dst, cnt, N_EDGES);
    scatter_kernel<<<N_EDGES / 8, 256, 0, stream>>>(x, src, dst, agg, N_EDGES);
    {
        dim3 grid(N_NODES / 16, HID / 64);   // 6250 x 2 ; 4 waves/block -> 8 out tiles
        sage_gemm_kernel<<<grid, 128, 0, stream>>>(x, agg, cnt, W1_l, b1_l, W1_r,
                                                   h, HID, /*relu=*/1);
    }

    // ---- layer 2 ----
    zero_f4<<<(zero_agg_f4 + 255) / 256, 256, 0, stream>>>((float4*)agg, zero_agg_f4);
    scatter_kernel<<<N_EDGES / 8, 256, 0, stream>>>(h, src, dst, agg, N_EDGES);
    {
        dim3 grid(N_NODES / 16, OUT_CH / 64); // 6250 x 1 ; 4 out tiles
        sage_gemm_kernel<<<grid, 128, 0, stream>>>(h, agg, cnt, W2_l, b2_l, W2_r,
                                                   out, OUT_CH, /*relu=*/0);
    }
}